// SRAPModule_24163486007644
// MI455X (gfx1250) — compile-verified
//
#include <hip/hip_runtime.h>
#include <hip/hip_bf16.h>

// ---------------------------------------------------------------------------
// Problem constants (match reference)
// ---------------------------------------------------------------------------
#define Hh   128
#define Tt   7
#define Bb   512
#define ENT  20000
#define SEQW (2 * Tt + 1)          // 15
#define MTOT (Tt * Bb)             // 3584
#define NCHUNK 5                   // entity chunks for logsumexp partials
#define CHUNK_COLS 4000            // 20000 / 5
#define CHUNK_PAIRS 125            // 4000 / 32  (16x32 wave tiles, exact)

typedef float v2f __attribute__((ext_vector_type(2)));
typedef float v8f __attribute__((ext_vector_type(8)));

// ---------------------------------------------------------------------------
// WMMA helper: one wave computes a 16x16 f32 tile of  C = A(MxK) @ W^T,
// with A row-major (lda) and W row-major (N x K, ldw), using
// V_WMMA_F32_16X16X4_F32 over the K loop.
//
// Operand layout per CDNA5 ISA 7.12.2 (wave32):
//   A frag : lanes 0-15 -> M = lane,     VGPR{0,1} = K {0,1};  lanes 16-31: K {2,3}
//   B frag : lanes      -> N = lane%16,  VGPR{0,1} = K {0,1} (half 0) / {2,3} (half 1)
//   C/D    : VGPR v, lane -> M = v + 8*(lane/16), N = lane%16
// ---------------------------------------------------------------------------
__device__ __forceinline__ v8f wmma_gemm_tile(const float* __restrict__ A, int lda,
                                              const float* __restrict__ W, int ldw,
                                              int m0, int n0, int K, int lane) {
  const int half = lane >> 4, lm = lane & 15;
  const float* arow = A + (size_t)(m0 + lm) * lda;
  const float* wrow = W + (size_t)(n0 + lm) * ldw;
  v8f acc = {0.f, 0.f, 0.f, 0.f, 0.f, 0.f, 0.f, 0.f};
  for (int k = 0; k < K; k += 4) {
    const int kk = k + 2 * half;
    v2f a = {arow[kk], arow[kk + 1]};
    v2f b = {wrow[kk], wrow[kk + 1]};
    acc = __builtin_amdgcn_wmma_f32_16x16x4_f32(false, a, false, b,
                                                (short)0, acc, false, false);
  }
  return acc;
}

__device__ __forceinline__ float hmax16(float v) {
  for (int m = 1; m < 16; m <<= 1) v = fmaxf(v, __shfl_xor(v, m, 16));
  return v;
}
__device__ __forceinline__ float hsum16(float v) {
  for (int m = 1; m < 16; m <<= 1) v += __shfl_xor(v, m, 16);
  return v;
}
__device__ __forceinline__ float hsum32(float v) {
  for (int m = 1; m < 32; m <<= 1) v += __shfl_xor(v, m, 32);
  return v;
}
__device__ __forceinline__ float sigmoidf(float x) { return 1.f / (1.f + expf(-x)); }

// ---------------------------------------------------------------------------
// Kernel: zero the scalar loss
// ---------------------------------------------------------------------------
__global__ void k_zero(float* loss) { *loss = 0.f; }

// ---------------------------------------------------------------------------
// Kernel: embedding gathers + fused bn scale; layout [t][b][j].
// Also initializes h0 = raw se[:,0,:].
// ---------------------------------------------------------------------------
__global__ void k_gather_bn(const int* __restrict__ seqs,
                            const float* __restrict__ sub_t,
                            const float* __restrict__ rel_t,
                            const float* __restrict__ obj_t,
                            const float* __restrict__ g1, const float* __restrict__ b1,
                            const float* __restrict__ g2, const float* __restrict__ b2,
                            float* __restrict__ st, float* __restrict__ rt,
                            float* __restrict__ tgt, float* __restrict__ h) {
  const int i = blockIdx.x * blockDim.x + threadIdx.x;
  if (i >= Tt * Bb * Hh) return;
  const int j = i % Hh;
  const int bt = i / Hh;
  const int b = bt % Bb;
  const int t = bt / Bb;
  const int* srow = seqs + b * SEQW;
  const int si = srow[2 * t], ri = srow[2 * t + 1], oi = srow[2 * t + 2];
  const float inv = rsqrtf(1.f + 1e-5f);
  const float sv = sub_t[(size_t)si * Hh + j];
  st[i]  = sv * (g1[j] * inv) + b1[j];
  rt[i]  = rel_t[(size_t)ri * Hh + j] * (g2[j] * inv) + b2[j];
  tgt[i] = obj_t[(size_t)oi * Hh + j];
  if (t == 0) h[b * Hh + j] = sv;
}

// ---------------------------------------------------------------------------
// Kernel: 4 fused 512x128x128 GEMMs (blockIdx.z selects which).
//   z=0: x  = st @ W1^T        -> xh[:, 0:128]
//   z=1: hh = h  @ W2w^T + W2b -> xh[:, 128:256]
//   z=2: u3 = h  @ W3^T        -> u3
//   z=3: u4 = rt @ W4w^T + W4b -> u4
// grid (32,1,4), block 256 (8 waves = 8 n-tiles).
// ---------------------------------------------------------------------------
__global__ void k_step_gemm4(const float* __restrict__ st_t, const float* __restrict__ h,
                             const float* __restrict__ rt_t,
                             const float* __restrict__ W1, const float* __restrict__ W2w,
                             const float* __restrict__ W2b, const float* __restrict__ W3,
                             const float* __restrict__ W4w, const float* __restrict__ W4b,
                             float* __restrict__ xh, float* __restrict__ u3,
                             float* __restrict__ u4) {
  const int wave = threadIdx.x >> 5, lane = threadIdx.x & 31;
  const int m0 = blockIdx.x * 16, n0 = wave * 16;
  const float* A; const float* W; const float* bias = nullptr;
  float* C; int ldc = Hh; int coff = 0;
  switch (blockIdx.z) {
    case 0:  A = st_t; W = W1;  C = xh; ldc = 2 * Hh;            break;
    case 1:  A = h;    W = W2w; C = xh; ldc = 2 * Hh; coff = Hh; bias = W2b; break;
    case 2:  A = h;    W = W3;  C = u3;                          break;
    default: A = rt_t; W = W4w; C = u4;              bias = W4b; break;
  }
  v8f acc = wmma_gemm_tile(A, Hh, W, Hh, m0, n0, Hh, lane);
  const int half = lane >> 4, lm = lane & 15;
  const float bv = bias ? bias[n0 + lm] : 0.f;
  for (int v = 0; v < 8; ++v)
    C[(size_t)(m0 + v + 8 * half) * ldc + coff + n0 + lm] = acc[v] + bv;
}

// ---------------------------------------------------------------------------
// Kernel: gate GEMM (K=256) + elementwise gate math.
//   g   = sigmoid(xh @ Gw^T + Gb)
//   op1 = tanh(g*hh + (1-g)*x)
//   op2 = sigmoid(u3*u4)  -> becomes h_next (written in place to h)
// grid (32), block 256.
// ---------------------------------------------------------------------------
__global__ void k_step_gate(const float* __restrict__ xh, const float* __restrict__ Gw,
                            const float* __restrict__ Gb, const float* __restrict__ u3,
                            const float* __restrict__ u4, float* __restrict__ op1,
                            float* __restrict__ h) {
  const int wave = threadIdx.x >> 5, lane = threadIdx.x & 31;
  const int m0 = blockIdx.x * 16, n0 = wave * 16;
  v8f acc = wmma_gemm_tile(xh, 2 * Hh, Gw, 2 * Hh, m0, n0, 2 * Hh, lane);
  const int half = lane >> 4, lm = lane & 15;
  for (int v = 0; v < 8; ++v) {
    const int m = m0 + v + 8 * half, n = n0 + lm;
    const float gg = sigmoidf(acc[v] + Gb[n]);
    const float xv = xh[(size_t)m * 2 * Hh + n];
    const float hv = xh[(size_t)m * 2 * Hh + Hh + n];
    op1[(size_t)m * Hh + n] = tanhf(gg * hv + (1.f - gg) * xv);
    h[(size_t)m * Hh + n]   = sigmoidf(u3[(size_t)m * Hh + n] * u4[(size_t)m * Hh + n]);
  }
}

// ---------------------------------------------------------------------------
// Kernel: output GEMMs.  z=0: xc = op1 @ W5^T ; z=1: hc = h @ W6w^T + W6b
// grid (32,1,2), block 256.
// ---------------------------------------------------------------------------
__global__ void k_step_outp(const float* __restrict__ op1, const float* __restrict__ h,
                            const float* __restrict__ W5, const float* __restrict__ W6w,
                            const float* __restrict__ W6b,
                            float* __restrict__ xc, float* __restrict__ hc) {
  const int wave = threadIdx.x >> 5, lane = threadIdx.x & 31;
  const int m0 = blockIdx.x * 16, n0 = wave * 16;
  const float* A = (blockIdx.z == 0) ? op1 : h;
  const float* W = (blockIdx.z == 0) ? W5 : W6w;
  float* C       = (blockIdx.z == 0) ? xc : hc;
  v8f acc = wmma_gemm_tile(A, Hh, W, Hh, m0, n0, Hh, lane);
  const int half = lane >> 4, lm = lane & 15;
  const float bv = (blockIdx.z == 0) ? 0.f : W6b[n0 + lm];
  for (int v = 0; v < 8; ++v)
    C[(size_t)(m0 + v + 8 * half) * Hh + n0 + lm] = acc[v] + bv;
}

// ---------------------------------------------------------------------------
// Kernel: complex-multiply combine + fused bn3 -> outputs[t]
// ---------------------------------------------------------------------------
__global__ void k_step_combine(const float* __restrict__ xc, const float* __restrict__ hc,
                               const float* __restrict__ g3, const float* __restrict__ b3,
                               float* __restrict__ out_t) {
  const int i = blockIdx.x * blockDim.x + threadIdx.x;   // over B*64
  if (i >= Bb * 64) return;
  const int b = i >> 6, j = i & 63;
  const float x1 = xc[(size_t)b * Hh + j],      x2 = xc[(size_t)b * Hh + 64 + j];
  const float h1 = hc[(size_t)b * Hh + j],      h2 = hc[(size_t)b * Hh + 64 + j];
  const float inv = rsqrtf(1.f + 1e-5f);
  out_t[(size_t)b * Hh + j]      = (x1 * h1 - x2 * h2) * (g3[j] * inv)      + b3[j];
  out_t[(size_t)b * Hh + 64 + j] = (x1 * h2 + x2 * h1) * (g3[64 + j] * inv) + b3[64 + j];
}

// ---------------------------------------------------------------------------
// Kernel: big logits GEMM (3584 x 20000, K=128) with fused online logsumexp.
// grid (224, 5); block 256 = 8 waves. Each wave computes 16x32 tiles (two
// accumulators sharing one A fragment -> 1 A-load + 2 B-loads per 2 WMMA),
// keeping running (max, sum-exp) per row; block-level merge in LDS; (m,s)
// partials stored per chunk. Logits never touch memory.
// ---------------------------------------------------------------------------
__global__ void k_logits_lse(const float* __restrict__ outs, const float* __restrict__ obj,
                             float* __restrict__ pm, float* __restrict__ ps) {
  __shared__ float smM[8][16];
  __shared__ float smS[8][16];
  const int wave = threadIdx.x >> 5, lane = threadIdx.x & 31;
  const int half = lane >> 4, lm = lane & 15;
  const int m0 = blockIdx.x * 16;
  const int nbase = blockIdx.y * CHUNK_COLS;
  const float* arow = outs + (size_t)(m0 + lm) * Hh;

  float mrun[8], srun[8];
  for (int v = 0; v < 8; ++v) { mrun[v] = -1e30f; srun[v] = 0.f; }

  for (int pair = wave; pair < CHUNK_PAIRS; pair += 8) {
    const int n0 = nbase + pair * 32;
    const float* wrow0 = obj + (size_t)(n0 + lm) * Hh;        // cols n0   .. n0+15
    const float* wrow1 = wrow0 + (size_t)16 * Hh;             // cols n0+16.. n0+31
    if (pair + 8 < CHUNK_PAIRS)                               // hint next pair into cache
      __builtin_prefetch(obj + (size_t)(n0 + 256 + lm) * Hh, 0, 0);
    v8f acc0 = {0.f, 0.f, 0.f, 0.f, 0.f, 0.f, 0.f, 0.f};
    v8f acc1 = {0.f, 0.f, 0.f, 0.f, 0.f, 0.f, 0.f, 0.f};
    for (int k = 0; k < Hh; k += 4) {
      const int kk = k + 2 * half;
      v2f a  = {arow[kk],  arow[kk + 1]};
      v2f b0 = {wrow0[kk], wrow0[kk + 1]};
      v2f b1 = {wrow1[kk], wrow1[kk + 1]};
      acc0 = __builtin_amdgcn_wmma_f32_16x16x4_f32(false, a, false, b0,
                                                   (short)0, acc0, false, false);
      acc1 = __builtin_amdgcn_wmma_f32_16x16x4_f32(false, a, false, b1,
                                                   (short)0, acc1, false, false);
    }
    // online (max, sum-exp) update per row; 16-lane butterflies make the
    // stats uniform within each half-group.
    for (int v = 0; v < 8; ++v) {
      const float mx   = fmaxf(hmax16(acc0[v]), hmax16(acc1[v]));
      const float mnew = fmaxf(mrun[v], mx);
      const float e    = hsum16(expf(acc0[v] - mnew)) + hsum16(expf(acc1[v] - mnew));
      srun[v] = srun[v] * expf(mrun[v] - mnew) + e;
      mrun[v] = mnew;
    }
  }
  if (lm == 0)
    for (int v = 0; v < 8; ++v) {
      smM[wave][v + 8 * half] = mrun[v];
      smS[wave][v + 8 * half] = srun[v];
    }
  __syncthreads();
  if (threadIdx.x < 16) {
    const int r = threadIdx.x;
    float m = -1e30f;
    for (int w = 0; w < 8; ++w) m = fmaxf(m, smM[w][r]);
    float s = 0.f;
    for (int w = 0; w < 8; ++w) s += smS[w][r] * expf(smM[w][r] - m);
    pm[(size_t)blockIdx.y * MTOT + m0 + r] = m;
    ps[(size_t)blockIdx.y * MTOT + m0 + r] = s;
  }
}

// ---------------------------------------------------------------------------
// Kernel: merge chunk partials, compute positive term, accumulate loss.
// One wave per row; grid 448 x 256 (8 waves/block).
// ---------------------------------------------------------------------------
__global__ void k_loss_finalize(const float* __restrict__ outs, const float* __restrict__ tgt,
                                const float* __restrict__ pm, const float* __restrict__ ps,
                                float* __restrict__ loss) {
  const int wave = threadIdx.x >> 5, lane = threadIdx.x & 31;
  const int row = blockIdx.x * 8 + wave;
  if (row >= MTOT) return;
  float p = 0.f;
  const float* o = outs + (size_t)row * Hh;
  const float* t = tgt + (size_t)row * Hh;
  for (int j = lane; j < Hh; j += 32) p += o[j] * t[j];
  p = hsum32(p);
  if (lane == 0) {
    float m = -1e30f;
    for (int c = 0; c < NCHUNK; ++c) m = fmaxf(m, pm[(size_t)c * MTOT + row]);
    float s = 0.f;
    for (int c = 0; c < NCHUNK; ++c) s += ps[(size_t)c * MTOT + row] * expf(pm[(size_t)c * MTOT + row] - m);
    atomicAdd(loss, m + logf(s) - p);
  }
}

// ---------------------------------------------------------------------------
// Host launcher
// ---------------------------------------------------------------------------
extern "C" void kernel_launch(void* const* d_in, const int* in_sizes, int n_in,
                              void* d_out, int out_size, void* d_ws, size_t ws_size,
                              hipStream_t stream) {
  const int*   seqs      = (const int*)d_in[0];
  const float* sub_table = (const float*)d_in[1];
  const float* rel_table = (const float*)d_in[2];
  const float* obj_table = (const float*)d_in[3];
  const float* W1  = (const float*)d_in[4];
  const float* W2w = (const float*)d_in[5];
  const float* W2b = (const float*)d_in[6];
  const float* W3  = (const float*)d_in[7];
  const float* W4w = (const float*)d_in[8];
  const float* W4b = (const float*)d_in[9];
  const float* W5  = (const float*)d_in[10];
  const float* W6w = (const float*)d_in[11];
  const float* W6b = (const float*)d_in[12];
  const float* Gw  = (const float*)d_in[13];
  const float* Gb  = (const float*)d_in[14];
  const float* g1  = (const float*)d_in[15];
  const float* b1  = (const float*)d_in[16];
  const float* g2  = (const float*)d_in[17];
  const float* b2  = (const float*)d_in[18];
  const float* g3  = (const float*)d_in[19];
  const float* b3  = (const float*)d_in[20];
  float* loss = (float*)d_out;

  // ---- workspace carve-up (floats) ----
  float* ws = (float*)d_ws;
  const size_t NTB = (size_t)Tt * Bb * Hh;   // 458752
  const size_t BH  = (size_t)Bb * Hh;        // 65536
  float* st   = ws;                 // [T][B][H] bn'd subject embeddings
  float* rt   = st  + NTB;          // [T][B][H] bn'd relation embeddings
  float* tgt  = rt  + NTB;          // [T][B][H] object (target) embeddings
  float* outs = tgt + NTB;          // [T][B][H] bn'd op3 outputs
  float* h    = outs + NTB;         // [B][H]   recurrent state
  float* xh   = h   + BH;           // [B][2H]  concat(x, hh)
  float* u3   = xh  + 2 * BH;       // [B][H]
  float* u4   = u3  + BH;           // [B][H]
  float* op1  = u4  + BH;           // [B][H]
  float* xc   = op1 + BH;           // [B][H]
  float* hc   = xc  + BH;           // [B][H]
  float* pm   = hc  + BH;           // [NCHUNK][MTOT] chunk max
  float* ps   = pm  + (size_t)NCHUNK * MTOT; // [NCHUNK][MTOT] chunk sum-exp

  k_zero<<<1, 1, 0, stream>>>(loss);

  const int gatherN = Tt * Bb * Hh;
  k_gather_bn<<<dim3((gatherN + 255) / 256), 256, 0, stream>>>(
      seqs, sub_table, rel_table, obj_table, g1, b1, g2, b2, st, rt, tgt, h);

  for (int t = 0; t < Tt; ++t) {
    const float* st_t = st + (size_t)t * BH;
    const float* rt_t = rt + (size_t)t * BH;
    float* out_t = outs + (size_t)t * BH;
    k_step_gemm4<<<dim3(Bb / 16, 1, 4), 256, 0, stream>>>(
        st_t, h, rt_t, W1, W2w, W2b, W3, W4w, W4b, xh, u3, u4);
    k_step_gate<<<dim3(Bb / 16), 256, 0, stream>>>(xh, Gw, Gb, u3, u4, op1, h);
    k_step_outp<<<dim3(Bb / 16, 1, 2), 256, 0, stream>>>(op1, h, W5, W6w, W6b, xc, hc);
    k_step_combine<<<dim3((Bb * 64 + 255) / 256), 256, 0, stream>>>(xc, hc, g3, b3, out_t);
  }

  k_logits_lse<<<dim3(MTOT / 16, NCHUNK), 256, 0, stream>>>(outs, obj_table, pm, ps);
  k_loss_finalize<<<dim3(MTOT / 8), 256, 0, stream>>>(outs, tgt, pm, ps, loss);
}